// GCN_52664888984064
// MI455X (gfx1250) — compile-verified
//
#include <hip/hip_runtime.h>
#include <hip/hip_bf16.h>

// ---------------------------------------------------------------------------
// GCN on MI455X (gfx1250).
//   h1 = relu(scatter(x @ W1) + b1); h2 = relu(scatter(h1 @ W2) + b2)
//   g  = meanpool(h2); out = relu(g @ Wf1 + bf1) @ Wf2 + bf2
// GEMMs use V_WMMA_F32_16X16X4_F32 (exact fp32; compute is not the bottleneck —
// the edge scatter is L2-bandwidth bound since 25.6MB of features << 192MB L2).
// W is staged in LDS in K-pair-major layout so each B fragment is one aligned
// ds_load_b64 straight into the even-aligned VGPR pair WMMA wants (no movs).
// ---------------------------------------------------------------------------

#define GN   50000
#define GE   800000
#define GNG  512
#define GD   128

typedef __attribute__((ext_vector_type(2))) float v2f;
typedef __attribute__((ext_vector_type(8))) float v8f;

// ---------------- small utility kernels ----------------

__global__ void k_fill(float* __restrict__ p, float v, long long n) {
    long long i = (long long)blockIdx.x * blockDim.x + threadIdx.x;
    if (i < n) p[i] = v;
}

__global__ void k_deg_accum(const int* __restrict__ dst, float* __restrict__ deg, int e_cnt) {
    int e = blockIdx.x * blockDim.x + threadIdx.x;
    if (e < e_cnt) atomicAdd(&deg[dst[e]], 1.0f);
}

__global__ void k_rsqrt_inplace(float* __restrict__ p, int n) {
    int i = blockIdx.x * blockDim.x + threadIdx.x;
    if (i < n) p[i] = rsqrtf(p[i]);
}

__global__ void k_norm(const float* __restrict__ dinv, const int* __restrict__ src,
                       const int* __restrict__ dst, float* __restrict__ norm,
                       int etot, int eorig) {
    int e = blockIdx.x * blockDim.x + threadIdx.x;
    if (e >= etot) return;
    if (e < eorig) norm[e] = dinv[src[e]] * dinv[dst[e]];
    else { float v = dinv[e - eorig]; norm[e] = v * v; }
}

// scatter-add:  out[dst] += T[src] * norm   (1 thread per (edge, feature))
__global__ void k_scatter(const float* __restrict__ T, const int* __restrict__ src,
                          const int* __restrict__ dst, const float* __restrict__ norm,
                          float* __restrict__ out, int etot, int eorig) {
    long long gid = (long long)blockIdx.x * blockDim.x + threadIdx.x;
    int e = (int)(gid >> 7);
    int f = (int)(gid & 127);
    if (e >= etot) return;
    int s, d;
    if (e < eorig) { s = src[e]; d = dst[e]; }
    else           { s = d = e - eorig; }
    atomicAdd(&out[(long long)d * GD + f], T[(long long)s * GD + f] * norm[e]);
}

__global__ void k_bias_relu(float* __restrict__ h, const float* __restrict__ b, long long total) {
    long long i = (long long)blockIdx.x * blockDim.x + threadIdx.x;
    if (i >= total) return;
    h[i] = fmaxf(h[i] + b[(int)(i & 127)], 0.0f);
}

__global__ void k_pool_sum(const float* __restrict__ h, const int* __restrict__ batch,
                           float* __restrict__ g, float* __restrict__ cnt, int n_nodes) {
    long long gid = (long long)blockIdx.x * blockDim.x + threadIdx.x;
    int nd = (int)(gid >> 7);
    int f  = (int)(gid & 127);
    if (nd >= n_nodes) return;
    int b = batch[nd];
    atomicAdd(&g[(long long)b * GD + f], h[(long long)nd * GD + f]);
    if (f == 0) atomicAdd(&cnt[b], 1.0f);
}

__global__ void k_pool_div(float* __restrict__ g, const float* __restrict__ cnt, int total) {
    int i = blockIdx.x * blockDim.x + threadIdx.x;
    if (i < total) g[i] /= fmaxf(cnt[i >> 7], 1.0f);
}

// out[512,2] = z[512,128] @ Wf2[128,2] + bf2
__global__ void k_fc2(const float* __restrict__ z, const float* __restrict__ Wf2,
                      const float* __restrict__ bf2, float* __restrict__ out) {
    int i = blockIdx.x * blockDim.x + threadIdx.x;
    if (i >= GNG * 2) return;
    int r = i >> 1, c = i & 1;
    float s = bf2[c];
    #pragma unroll 8
    for (int k = 0; k < GD; ++k) s += z[r * GD + k] * Wf2[k * 2 + c];
    out[i] = s;
}

// ---------------- WMMA GEMM:  C[M x 128] = A[M x 128] @ W[128 x 128] ----------------
// One wave computes one 16-row M tile across all 8 N tiles (8 f32 accumulators).
// W staged in LDS (64KB) as K-pair-major v2f: sW2[(k>>1)*128 + col] = {W[k][col],
// W[k+1][col]}, so every B fragment is one aligned ds_load_b64. M must be a
// multiple of 16 (50000 and 512 both are). Optional fused bias + relu.

__global__ void __launch_bounds__(256)
k_gemm128_wmma(const float* __restrict__ A, const float* __restrict__ W,
               const float* __restrict__ bias, float* __restrict__ C,
               int M, int apply_relu) {
    __shared__ __align__(16) float sW[GD * GD];

    const int tid = threadIdx.x;                    // 0..255 (8 waves, wave32)
    // cooperative load of W, repacked into K-pair-major layout.
    // float4 chunk = row k, cols c..c+3 -> sW[(k>>1)*256 + 2*col + (k&1)]
    const float4* Wv = (const float4*)W;
    #pragma unroll
    for (int i = 0; i < 16; ++i) {
        int flat = (tid + i * 256) * 4;             // flat float index into W
        int k    = flat >> 7;
        int c    = flat & 127;
        float4 w = Wv[tid + i * 256];
        int base = ((k >> 1) << 8) + (k & 1) + 2 * c;
        sW[base + 0] = w.x;
        sW[base + 2] = w.y;
        sW[base + 4] = w.z;
        sW[base + 6] = w.w;
    }
    __syncthreads();

    const int wave  = tid >> 5;
    const int lane  = tid & 31;
    const int tileM = blockIdx.x * 8 + wave;
    if (tileM * 16 >= M) return;                    // wave-uniform: EXEC all-1s for WMMA

    const int half = lane >> 4;                     // 0: K pair {k,k+1}; 1: {k+2,k+3}
    const int l15  = lane & 15;
    const int arow = tileM * 16 + l15;              // A-frag row (per ISA 16x4 layout)

    const v2f* sW2 = (const v2f*)sW;                // [64][128] K-pair-major

    v8f acc[8] = {};

    for (int k0 = 0; k0 < GD; k0 += 4) {
        const int ka = k0 + (half ? 2 : 0);
        const int kp = (k0 >> 1) + half;            // K-pair index for this lane
        v2f a;
        a.x = A[(long long)arow * GD + ka];         // contiguous -> global_load_b64
        a.y = A[(long long)arow * GD + ka + 1];
        #pragma unroll
        for (int n = 0; n < 8; ++n) {
            v2f b = sW2[kp * GD + n * 16 + l15];    // one ds_load_b64, even-aligned
            acc[n] = __builtin_amdgcn_wmma_f32_16x16x4_f32(
                false, a, false, b, (short)0, acc[n], false, false);
        }
    }

    // C/D 16x16 f32 layout: VGPR i -> row (i + 8*half), col = n*16 + l15
    #pragma unroll
    for (int n = 0; n < 8; ++n) {
        #pragma unroll
        for (int i = 0; i < 8; ++i) {
            int row = tileM * 16 + i + (half ? 8 : 0);
            int col = n * 16 + l15;
            float v = acc[n][i];
            if (bias) v += bias[col];
            if (apply_relu) v = fmaxf(v, 0.0f);
            C[(long long)row * GD + col] = v;
        }
    }
}

// ---------------- host-side orchestration ----------------

extern "C" void kernel_launch(void* const* d_in, const int* in_sizes, int n_in,
                              void* d_out, int out_size, void* d_ws, size_t ws_size,
                              hipStream_t stream) {
    (void)in_sizes; (void)n_in; (void)out_size; (void)ws_size;

    const float* x    = (const float*)d_in[0];
    const int*   edge = (const int*)  d_in[1];   // [2, E] flattened
    const int*   batch= (const int*)  d_in[2];
    const float* W1   = (const float*)d_in[3];
    const float* b1   = (const float*)d_in[4];
    const float* W2   = (const float*)d_in[5];
    const float* b2   = (const float*)d_in[6];
    const float* Wf1  = (const float*)d_in[7];
    const float* bf1  = (const float*)d_in[8];
    const float* Wf2  = (const float*)d_in[9];
    const float* bf2  = (const float*)d_in[10];
    float* out = (float*)d_out;

    const int* src = edge;
    const int* dst = edge + GE;
    const int  ETOT = GE + GN;

    // workspace layout (floats)
    float* ws   = (float*)d_ws;
    float* bufA = ws;                               // [N,128] GEMM output
    float* bufB = bufA + (long long)GN * GD;        // [N,128] scatter accum / activations
    float* dinv = bufB + (long long)GN * GD;        // [N]  deg -> rsqrt(deg)
    float* norm = dinv + GN;                        // [E+N]
    float* g    = norm + ETOT;                      // [NG,128]
    float* cnt  = g + GNG * GD;                     // [NG]
    float* z    = cnt + GNG;                        // [NG,128]

    const int T = 256;
    const long long NF = (long long)GN * GD;
    dim3 blk(T);
    auto nb = [](long long n, int t) { return (unsigned)((n + t - 1) / t); };

    // ---- normalization coefficients ----
    k_fill<<<nb(GN, T), blk, 0, stream>>>(dinv, 1.0f, GN);          // self-loop deg
    k_deg_accum<<<nb(GE, T), blk, 0, stream>>>(dst, dinv, GE);
    k_rsqrt_inplace<<<nb(GN, T), blk, 0, stream>>>(dinv, GN);
    k_norm<<<nb(ETOT, T), blk, 0, stream>>>(dinv, src, dst, norm, ETOT, GE);

    const unsigned gemm_blocks_big = nb(GN / 16, 8);                // 3125 tiles -> 391 blocks
    const unsigned gemm_blocks_sm  = nb(GNG / 16, 8);               // 32 tiles   -> 4 blocks

    // ---- layer 1 ----
    k_gemm128_wmma<<<gemm_blocks_big, blk, 0, stream>>>(x, W1, nullptr, bufA, GN, 0);
    k_fill<<<nb(NF, T), blk, 0, stream>>>(bufB, 0.0f, NF);
    k_scatter<<<nb((long long)ETOT * GD, T), blk, 0, stream>>>(bufA, src, dst, norm, bufB, ETOT, GE);
    k_bias_relu<<<nb(NF, T), blk, 0, stream>>>(bufB, b1, NF);

    // ---- layer 2 ----
    k_gemm128_wmma<<<gemm_blocks_big, blk, 0, stream>>>(bufB, W2, nullptr, bufA, GN, 0);
    k_fill<<<nb(NF, T), blk, 0, stream>>>(bufB, 0.0f, NF);
    k_scatter<<<nb((long long)ETOT * GD, T), blk, 0, stream>>>(bufA, src, dst, norm, bufB, ETOT, GE);
    k_bias_relu<<<nb(NF, T), blk, 0, stream>>>(bufB, b2, NF);

    // ---- global mean pool ----
    k_fill<<<nb(GNG * GD, T), blk, 0, stream>>>(g, 0.0f, GNG * GD);
    k_fill<<<nb(GNG, T), blk, 0, stream>>>(cnt, 0.0f, GNG);
    k_pool_sum<<<nb((long long)GN * GD, T), blk, 0, stream>>>(bufB, batch, g, cnt, GN);
    k_pool_div<<<nb(GNG * GD, T), blk, 0, stream>>>(g, cnt, GNG * GD);

    // ---- MLP head ----
    k_gemm128_wmma<<<gemm_blocks_sm, blk, 0, stream>>>(g, Wf1, bf1, z, GNG, 1);
    k_fc2<<<nb(GNG * 2, T), blk, 0, stream>>>(z, Wf2, bf2, out);
}